// TripletLoss_72387378806878
// MI455X (gfx1250) — compile-verified
//
#include <hip/hip_runtime.h>

typedef __attribute__((ext_vector_type(2))) float v2f;
typedef __attribute__((ext_vector_type(8))) float v8f;
typedef int v4i_vs __attribute__((vector_size(4 * sizeof(int))));  // matches builtin param

#define N_PTS   8192
#define DIM     256
#define MARGIN  0.3f
#define EPS_D2  1e-12f
#define APAD    260   // padded LDS row stride (floats): 260 % 64 == 4 -> conflict-free frags

#if defined(__has_builtin)
#if __has_builtin(__builtin_amdgcn_global_load_async_to_lds_b128)
#define TL_ASYNC_BUILTIN 1
#endif
#if __has_builtin(__builtin_amdgcn_s_wait_asynccnt)
#define TL_WAIT_BUILTIN 1
#endif
#endif

// 16-byte global -> LDS async copy (ASYNCcnt-tracked, no VGPR data path).
__device__ __forceinline__ void tl_async_cp16(const float* gp, unsigned lds_byte_off) {
#ifdef TL_ASYNC_BUILTIN
  __builtin_amdgcn_global_load_async_to_lds_b128(
      (__attribute__((address_space(1))) v4i_vs*)(uintptr_t)gp,
      (__attribute__((address_space(3))) v4i_vs*)(uintptr_t)lds_byte_off,
      0, 0);
#else
  asm volatile("global_load_async_to_lds_b128 %0, %1, off"
               :: "v"(lds_byte_off), "v"(gp) : "memory");
#endif
}

__device__ __forceinline__ void tl_async_wait0() {
#ifdef TL_WAIT_BUILTIN
  __builtin_amdgcn_s_wait_asynccnt(0);
#else
  asm volatile("s_wait_asynccnt 0x0" ::: "memory");
#endif
}

// ---------------------------------------------------------------------------
// Kernel 1: row squared norms + init mining buffers (int-bit ordered atomics).
// ---------------------------------------------------------------------------
__global__ __launch_bounds__(256) void tl_prep_kernel(
    const float* __restrict__ x, float* __restrict__ sq,
    int* __restrict__ ap, int* __restrict__ an) {
  const int wave = threadIdx.x >> 5;
  const int lane = threadIdx.x & 31;
  const int row  = blockIdx.x * 8 + wave;
  const float* xr = x + (size_t)row * DIM;
  float s = 0.0f;
#pragma unroll
  for (int t = 0; t < DIM; t += 32) {
    float v = xr[t + lane];
    s = fmaf(v, v, s);
  }
#pragma unroll
  for (int off = 16; off >= 1; off >>= 1) s += __shfl_xor(s, off, 32);
  if (lane == 0) {
    sq[row] = s;
    ap[row] = 0;           // 0.0f bits
    an[row] = 0x7F800000;  // +inf bits
  }
}

// ---------------------------------------------------------------------------
// Kernel 2: fused X*X^T (fp32 WMMA 16x16x4) + dist + batch-hard mining,
// with async-to-LDS double-buffered tiles and dual accumulator chains.
//   Block = 128 threads (4 waves). Rows [i0, i0+64), columns [j0b, j0b+512).
//   As: 64 x 256 staged once. Bs: 2 x (16 x 256) double buffer.
// Fragment layouts (ISA 7.12.2, 32-bit, wave32):
//   A 16x4 : h=lane>>4, m=lane&15 ; VGPR0/1 = row(i0+m) cols [k+2h, k+2h+1]
//   B 4x16 : same striping on the column-row (j0+n)
//   C 16x16: lane n=lane&15 ; c[v] = G[i0 + v + 8h][j0 + n]
// ---------------------------------------------------------------------------
__global__ __launch_bounds__(128) void tl_mine_kernel(
    const float* __restrict__ x, const int* __restrict__ lab,
    const float* __restrict__ sq, int* __restrict__ ap, int* __restrict__ an) {
  __shared__ float As[64 * APAD];       // 66,560 B
  __shared__ float Bs[2 * 16 * APAD];   // 33,280 B

  const int t    = threadIdx.x;         // 0..127
  const int wave = t >> 5;              // 0..3
  const int lane = t & 31;
  const int half = lane >> 4;           // 0/1
  const int nl   = lane & 15;
  const int iblk = blockIdx.x * 64;     // block row base
  const int i0   = iblk + wave * 16;    // wave row base
  const int j0b  = blockIdx.y * 512;    // column strip base

  const unsigned As_off = (unsigned)(uintptr_t)As;
  const unsigned Bs_off = (unsigned)(uintptr_t)Bs;

  // ---- stage A block (64 rows x 256 f32 = 4096 x 16B chunks, 32/thread) ----
#pragma unroll
  for (int c = 0; c < 32; ++c) {
    const int chunk = t + 128 * c;
    const int row  = chunk >> 6;        // 64 chunks per row
    const int col4 = chunk & 63;
    tl_async_cp16(x + (size_t)(iblk + row) * DIM + col4 * 4,
                  As_off + (unsigned)((row * APAD + col4 * 4) * 4));
  }
  // ---- stage B tile 0 (16 rows = 1024 chunks, 8/thread) ----
#pragma unroll
  for (int c = 0; c < 8; ++c) {
    const int chunk = t + 128 * c;
    const int row  = chunk >> 6;
    const int col4 = chunk & 63;
    tl_async_cp16(x + (size_t)(j0b + row) * DIM + col4 * 4,
                  Bs_off + (unsigned)((row * APAD + col4 * 4) * 4));
  }
  tl_async_wait0();
  __syncthreads();

  int   li[8];
  float sqi[8];
#pragma unroll
  for (int v = 0; v < 8; ++v) {
    const int r = i0 + half * 8 + v;
    li[v]  = lab[r];
    sqi[v] = sq[r];
  }

  float apm[8], anm[8];
#pragma unroll
  for (int v = 0; v < 8; ++v) { apm[v] = -INFINITY; anm[v] = INFINITY; }

  const float* __restrict__ aB = As + (wave * 16 + nl) * APAD + 2 * half;

  int cur = 0;
  for (int jt = 0; jt < 32; ++jt) {
    // Prefetch next B tile into the other buffer while we compute this one.
    if (jt + 1 < 32) {
      const int jn = j0b + (jt + 1) * 16;
      const int nb = cur ^ 1;
#pragma unroll
      for (int c = 0; c < 8; ++c) {
        const int chunk = t + 128 * c;
        const int row  = chunk >> 6;
        const int col4 = chunk & 63;
        tl_async_cp16(x + (size_t)(jn + row) * DIM + col4 * 4,
                      Bs_off + (unsigned)(((nb * 16 + row) * APAD + col4 * 4) * 4));
      }
    }

    const int   j   = j0b + jt * 16 + nl;
    const int   lj  = lab[j];
    const float sqj = sq[j];
    const float* __restrict__ bB = Bs + (cur * 16 + nl) * APAD + 2 * half;

    // Two independent accumulator chains (even/odd k-steps) for WMMA ILP.
    v8f c0 = {}, c1 = {};
#pragma unroll
    for (int k = 0; k < DIM; k += 8) {
      v2f a0 = *(const v2f*)(aB + k);
      v2f b0 = *(const v2f*)(bB + k);
      c0 = __builtin_amdgcn_wmma_f32_16x16x4_f32(
               false, a0, false, b0, (short)0, c0, false, false);
      v2f a1 = *(const v2f*)(aB + k + 4);
      v2f b1 = *(const v2f*)(bB + k + 4);
      c1 = __builtin_amdgcn_wmma_f32_16x16x4_f32(
               false, a1, false, b1, (short)0, c1, false, false);
    }

#pragma unroll
    for (int v = 0; v < 8; ++v) {
      const float g  = c0[v] + c1[v];
      const float d2 = sqi[v] + sqj - 2.0f * g;
      const float d  = sqrtf(fmaxf(d2, EPS_D2));
      const bool pos = (li[v] == lj);
      apm[v] = fmaxf(apm[v], pos ? d : -INFINITY);
      anm[v] = fminf(anm[v], pos ? INFINITY : d);
    }

    // Single sync point: my async copies landed + everyone done reading cur.
    tl_async_wait0();
    __syncthreads();
    cur ^= 1;
  }

  // Reduce across the 16 lanes sharing each row m = v + 8*half.
#pragma unroll
  for (int v = 0; v < 8; ++v) {
#pragma unroll
    for (int off = 8; off >= 1; off >>= 1) {
      apm[v] = fmaxf(apm[v], __shfl_xor(apm[v], off, 16));
      anm[v] = fminf(anm[v], __shfl_xor(anm[v], off, 16));
    }
  }

  if (nl == 0) {
#pragma unroll
    for (int v = 0; v < 8; ++v) {
      const int r = i0 + half * 8 + v;
      atomicMax(&ap[r], __float_as_int(apm[v]));  // candidates: -inf or >= 0
      atomicMin(&an[r], __float_as_int(anm[v]));  // candidates: +inf or >= 0
    }
  }
}

// ---------------------------------------------------------------------------
// Kernel 3: loss = mean(relu(ap - an + margin))
// ---------------------------------------------------------------------------
__global__ __launch_bounds__(256) void tl_loss_kernel(
    const float* __restrict__ ap, const float* __restrict__ an,
    float* __restrict__ out) {
  __shared__ float red[256];
  float s = 0.0f;
  for (int r = threadIdx.x; r < N_PTS; r += 256)
    s += fmaxf(ap[r] - an[r] + MARGIN, 0.0f);
  red[threadIdx.x] = s;
  __syncthreads();
#pragma unroll
  for (int off = 128; off >= 1; off >>= 1) {
    if (threadIdx.x < off) red[threadIdx.x] += red[threadIdx.x + off];
    __syncthreads();
  }
  if (threadIdx.x == 0) out[0] = red[0] * (1.0f / (float)N_PTS);
}

extern "C" void kernel_launch(void* const* d_in, const int* in_sizes, int n_in,
                              void* d_out, int out_size, void* d_ws, size_t ws_size,
                              hipStream_t stream) {
  (void)in_sizes; (void)n_in; (void)out_size; (void)ws_size;
  const float* x   = (const float*)d_in[0];
  const int*   lab = (const int*)d_in[1];

  float* sq = (float*)d_ws;
  int*   ap = (int*)((char*)d_ws + (size_t)N_PTS * 4);
  int*   an = (int*)((char*)d_ws + (size_t)N_PTS * 8);

  tl_prep_kernel<<<N_PTS / 8, 256, 0, stream>>>(x, sq, ap, an);

  dim3 grid(N_PTS / 64, N_PTS / 512);  // 128 x 16 blocks, 4 waves each
  tl_mine_kernel<<<grid, 128, 0, stream>>>(x, lab, sq, ap, an);

  tl_loss_kernel<<<1, 256, 0, stream>>>((const float*)ap, (const float*)an,
                                        (float*)d_out);
}